// SpatialConv2d_8409545965922
// MI455X (gfx1250) — compile-verified
//
#include <hip/hip_runtime.h>
#include <math.h>

typedef __attribute__((ext_vector_type(2))) float v2f;
typedef __attribute__((ext_vector_type(8))) float v8f;

#define CIN   64
#define COUT  128
#define IMH   32
#define IMW   32
#define KTOT  576          // CIN * 9
#define NPIX  1024         // 32*32
#define NCH_X 69           // CIN + 5
#define BSTR  580          // padded LDS row stride (mod-64-bank offset = 4)

// ---------------------------------------------------------------------------
// Kernel 1: fused implicit-GEMM 3x3 conv (fp32 WMMA 16x16x4) + L1 shape
// distance + shape-attended features (out ch 0..127) + per-pixel q reduction.
// One workgroup = (b, h, 16-pixel strip); 8 waves = 8 M-tiles of 16 channels.
// B-panel is pre-expanded (im2col) into LDS so the 144-step K-loop has zero
// address VALU: ds_load_b64 + global_load_b64 + v_wmma per step.
// ---------------------------------------------------------------------------
__global__ __launch_bounds__(256) void spatialconv_wmma_kernel(
    const float* __restrict__ x,        // (B, 69, 32, 32)
    const float* __restrict__ conv_w,   // (128, 576)
    const float* __restrict__ conv_b,   // (128,)
    const float* __restrict__ conv2_w,  // (128,)
    const float* __restrict__ conv2_b,  // (1,)
    const float* __restrict__ shapes,   // (128, 45)
    float* __restrict__ out,            // (B, 133, 32, 32)
    float* __restrict__ qws)            // (B, 32, 32)
{
    __shared__ float ldsB[16 * BSTR];        // im2col panel  37120 B
    __shared__ float lds_sk[COUT * 45];      // shape kernels 23040 B
    __shared__ float lds_sw[45 * 16];        // shape windows  2880 B
    __shared__ float lds_part[256];          // q partials     1024 B

    const int bid = blockIdx.x;
    const int wt  = bid & 1;
    const int h   = (bid >> 1) & 31;
    const int b   = bid >> 6;
    const int w0  = wt * 16;

    const int tid  = threadIdx.x;
    const int lane = tid & 31;
    const int wv   = tid >> 5;       // wave id 0..7
    const int pix  = lane & 15;      // N column / A row
    const int lh   = lane >> 4;      // lane half -> K pair select

    const float* xb = x + (size_t)b * NCH_X * NPIX;

    // ---- async-stage shape kernels: contiguous 23040 B -> LDS ------------
    {
        unsigned lbase = (unsigned)(size_t)(void*)lds_sk;
        for (int idx = tid; idx < (COUT * 45) / 4; idx += 256) {
            unsigned loff = lbase + idx * 16u;
            const float* gaddr = shapes + idx * 4;
            asm volatile("global_load_async_to_lds_b128 %0, %1, off"
                         :: "v"(loff), "v"(gaddr) : "memory");
        }
    }

    // ---- build im2col B panel: ldsB[pix][t] = patch(c,i,j+pix) -----------
    for (int idx = tid; idx < 16 * KTOT; idx += 256) {
        int px = idx / KTOT;
        int t  = idx - px * KTOT;            // c*9 + i*3 + j
        int c  = t / 9,  p = t - c * 9;
        int i  = (p >= 6) ? 2 : ((p >= 3) ? 1 : 0);
        int j  = p - 3 * i;
        int hh = h - 1 + i, ww = w0 - 1 + j + px;
        float v = 0.0f;                      // zero padding (conv)
        if ((unsigned)hh < (unsigned)IMH && (unsigned)ww < (unsigned)IMW)
            v = xb[(c * IMH + hh) * IMW + ww];
        ldsB[px * BSTR + t] = v;
    }

    // ---- shape windows (edge pad, center-relative for ch 0,1) ------------
    for (int idx = tid; idx < 45 * 16; idx += 256) {
        int t  = idx >> 4, px = idx & 15;
        int c5 = t / 9,    p  = t % 9;
        int i  = (p >= 6) ? 2 : ((p >= 3) ? 1 : 0);
        int j  = p - 3 * i;
        int hh = min(max(h - 1 + i, 0), IMH - 1);
        int ww = min(max(w0 - 1 + j + px, 0), IMW - 1);
        float v = xb[((CIN + c5) * IMH + hh) * IMW + ww];
        if (c5 < 2) v -= xb[((CIN + c5) * IMH + h) * IMW + (w0 + px)];
        lds_sw[idx] = v;
    }

    asm volatile("s_wait_asynccnt 0x0" ::: "memory");
    __syncthreads();

    // ---- implicit GEMM: 144 x V_WMMA_F32_16X16X4_F32, zero-VALU loop -----
    const int m0 = wv * 16;
    const float* wptr = conv_w + (size_t)(m0 + pix) * KTOT + 2 * lh;
    const float* bptr = ldsB + pix * BSTR + 2 * lh;
    __builtin_prefetch(wptr, 0, 1);          // global_prefetch_b8

    v8f acc = {};
#pragma unroll
    for (int k = 0; k < KTOT; k += 4) {
        v2f a  = *(const v2f*)(wptr + k);    // global_load_b64, imm offset
        v2f bf = *(const v2f*)(bptr + k);    // ds_load_b64,     imm offset
        acc = __builtin_amdgcn_wmma_f32_16x16x4_f32(
            false, a, false, bf, (short)0, acc, false, false);
    }

    // ---- L1 shape distance ----------------------------------------------
    float sdv[8];
#pragma unroll
    for (int r = 0; r < 8; ++r) sdv[r] = 0.0f;
    const int mbase = m0 + 8 * lh;
    for (int t = 0; t < 45; ++t) {
        float swv = lds_sw[t * 16 + pix];
#pragma unroll
        for (int r = 0; r < 8; ++r)
            sdv[r] += fabsf(swv - lds_sk[(mbase + r) * 45 + t]);
    }

    // ---- saf = relu(conv / (sd + 1)), store + q partial ------------------
    float qpart = 0.0f;
    const size_t obase = (((size_t)b * 133) * IMH + h) * IMW + w0 + pix;
#pragma unroll
    for (int r = 0; r < 8; ++r) {
        int m = mbase + r;
        float v = (acc[r] + conv_b[m]) / (sdv[r] + 1.0f);
        v = fmaxf(v, 0.0f);
        out[obase + (size_t)m * NPIX] = v;
        qpart += v * conv2_w[m];
    }

    lds_part[wv * 32 + lane] = qpart;        // == (2*wv + lh)*16 + pix
    __syncthreads();
    if (tid < 16) {
        float s = 0.0f;
#pragma unroll
        for (int idx = 0; idx < 16; ++idx) s += lds_part[idx * 16 + tid];
        qws[((size_t)b * IMH + h) * IMW + w0 + tid] =
            (s + conv2_b[0]) * 0.08838834764831845f;   // 1/sqrt(128)
    }
}

// ---------------------------------------------------------------------------
// Kernel 2: per-pixel 3x3 softmax of q (zero padded) + weighted statistics
// (means, var, covar) over the 5 shape channels (edge padded) -> ch 128..132.
// ---------------------------------------------------------------------------
__global__ __launch_bounds__(256) void spatialconv_agg_kernel(
    const float* __restrict__ x,
    const float* __restrict__ qws,
    float* __restrict__ out)
{
    int gid = blockIdx.x * 256 + threadIdx.x;
    if (gid >= 16 * NPIX) return;
    int w = gid & 31, h = (gid >> 5) & 31, b = gid >> 10;

    const float* xb = x + (size_t)b * NCH_X * NPIX;
    const float* qb = qws + (size_t)b * NPIX;

    float qt[9];
    float mx = -3.0e38f;
#pragma unroll
    for (int p = 0; p < 9; ++p) {
        int i = p / 3, j = p % 3;
        int hh = h - 1 + i, ww = w - 1 + j;
        float v = 0.0f;   // zero padding (unfold default 'constant')
        if ((unsigned)hh < (unsigned)IMH && (unsigned)ww < (unsigned)IMW)
            v = qb[hh * IMW + ww];
        qt[p] = v;
        mx = fmaxf(mx, v);
    }
    float s = 0.0f;
#pragma unroll
    for (int p = 0; p < 9; ++p) { qt[p] = __expf(qt[p] - mx); s += qt[p]; }
    float inv = 1.0f / s;

    float u0[9], u1[9];
    float m0 = 0.f, m1 = 0.f, a0s = 0.f, a1s = 0.f, c1s = 0.f;
#pragma unroll
    for (int p = 0; p < 9; ++p) {
        qt[p] *= inv;
        int i = p / 3, j = p % 3;
        int hh = min(max(h - 1 + i, 0), IMH - 1);
        int ww = min(max(w - 1 + j, 0), IMW - 1);
        int base = hh * IMW + ww;
        float a = xb[(CIN + 0) * NPIX + base];
        float c = xb[(CIN + 1) * NPIX + base];
        u0[p] = a;  u1[p] = c;
        m0  += a * qt[p];
        m1  += c * qt[p];
        a0s += xb[(CIN + 2) * NPIX + base] * qt[p];
        a1s += xb[(CIN + 3) * NPIX + base] * qt[p];
        c1s += xb[(CIN + 4) * NPIX + base] * qt[p];
    }
    float v20 = 0.f, v21 = 0.f, c2s = 0.f;
#pragma unroll
    for (int p = 0; p < 9; ++p) {
        float d0 = u0[p] - m0, d1 = u1[p] - m1;
        v20 += d0 * d0 * qt[p];
        v21 += d1 * d1 * qt[p];
        c2s += d0 * d1 * qt[p];
    }

    size_t ob = ((size_t)b * 133) * NPIX + (size_t)h * IMW + w;
    out[ob + (size_t)128 * NPIX] = m0;
    out[ob + (size_t)129 * NPIX] = m1;
    out[ob + (size_t)130 * NPIX] = a0s + v20;
    out[ob + (size_t)131 * NPIX] = a1s + v21;
    out[ob + (size_t)132 * NPIX] = c1s + c2s;
}

// ---------------------------------------------------------------------------
extern "C" void kernel_launch(void* const* d_in, const int* in_sizes, int n_in,
                              void* d_out, int out_size, void* d_ws, size_t ws_size,
                              hipStream_t stream)
{
    const float* x        = (const float*)d_in[0];
    const float* conv_w   = (const float*)d_in[1];
    const float* conv_b   = (const float*)d_in[2];
    const float* conv2_w  = (const float*)d_in[3];
    const float* conv2_b  = (const float*)d_in[4];
    const float* shapes   = (const float*)d_in[5];
    float* out = (float*)d_out;
    float* qws = (float*)d_ws;   // 16*1024 floats = 64 KB workspace

    // 16 batches * 32 rows * 2 strips = 1024 workgroups of 8 waves
    spatialconv_wmma_kernel<<<1024, 256, 0, stream>>>(
        x, conv_w, conv_b, conv2_w, conv2_b, shapes, out, qws);

    spatialconv_agg_kernel<<<(16 * NPIX + 255) / 256, 256, 0, stream>>>(
        x, qws, out);
}